// DigitCaps_34866544509329
// MI455X (gfx1250) — compile-verified
//
#include <hip/hip_runtime.h>
#include <hip/hip_bf16.h>

// ---------------------------------------------------------------------------
// DigitCaps (capsule routing) for MI455X / gfx1250.
//   x      : [B=32, J=8192, Din=8]   fp32
//   weight : [J, 8, 160]             fp32
//   biases : [10, 16] = [160]        fp32
//   out    : [B, 10, 16] = [B, 160]  fp32
//
// Pose projection uses V_WMMA_F32_16X16X4_F32 (exact fp32 matrix path).
// Routing is 3 deterministic streaming passes over u (no atomics).
// ---------------------------------------------------------------------------

typedef __attribute__((ext_vector_type(2))) float v2f;
typedef __attribute__((ext_vector_type(8))) float v8f;

#define NUM_CAPS 10
#define CAP_LEN  16
#define KTOT     160              // NUM_CAPS * CAP_LEN
#define BATCH    32
#define DIN      8
#define SQ_EPS   1e-20f

// ---------------------------------------------------------------------------
// Kernel 1: pose projection. One workgroup (10 waves) per j.
// Wave w computes the 16-wide N-tile of capsule w for both 16-row M-tiles:
//   u[b, j, w*16 + n] = sum_{k=0..7} x[b, j, k] * W[j, k, w*16 + n]
// via two chained V_WMMA_F32_16X16X4_F32 per M-tile.
// ---------------------------------------------------------------------------
__global__ __launch_bounds__(320) void pose_wmma_kernel(
    const float* __restrict__ x,       // [B, J, 8]
    const float* __restrict__ weight,  // [J, 8, 160]
    float* __restrict__ u,             // [B, J, 160]
    int J) {
  const int j    = blockIdx.x;
  const int wave = threadIdx.x >> 5;   // capsule index 0..9
  const int lane = threadIdx.x & 31;
  const int col  = lane & 15;          // A: M index / B,D: N index
  const int grp  = lane >> 4;          // lane-group selects K pairs / M half

  // ---- B fragments: W[j, k, wave*16 + col], VGPR v -> k = grp*2 + v (+4*s)
  const float* Wj = weight + (size_t)j * (DIN * KTOT);
  const int    nglob = wave * CAP_LEN + col;
  v2f bf0, bf1;
  bf0.x = Wj[(grp * 2 + 0) * KTOT + nglob];
  bf0.y = Wj[(grp * 2 + 1) * KTOT + nglob];
  bf1.x = Wj[(grp * 2 + 4) * KTOT + nglob];
  bf1.y = Wj[(grp * 2 + 5) * KTOT + nglob];

#pragma unroll
  for (int mt = 0; mt < 2; ++mt) {     // M-tiles: batches 0..15 and 16..31
    // ---- A fragments: x[mt*16 + col, j, k], VGPR v -> k = grp*2 + v (+4*s)
    const float* xr = x + ((size_t)(mt * 16 + col) * J + j) * DIN;
    v2f af0, af1;
    af0.x = xr[grp * 2 + 0];
    af0.y = xr[grp * 2 + 1];
    af1.x = xr[grp * 2 + 4];
    af1.y = xr[grp * 2 + 5];

    v8f acc = {};
    acc = __builtin_amdgcn_wmma_f32_16x16x4_f32(false, af0, false, bf0,
                                                (short)0, acc, false, false);
    acc = __builtin_amdgcn_wmma_f32_16x16x4_f32(false, af1, false, bf1,
                                                (short)0, acc, false, false);

    // ---- D layout: lane col = N, VGPR v -> M = v + 8*grp
#pragma unroll
    for (int v = 0; v < 8; ++v) {
      const int brow = mt * 16 + v + 8 * grp;         // batch index 0..31
      u[((size_t)brow * J + j) * KTOT + nglob] = acc[v];
    }
  }
}

// ---------------------------------------------------------------------------
// Kernel 2: routing logits. Thread per (b, j):
//   a[n]   = sum_c u[b,j,n,c] * v_prev[b,n,c]
//   b_new  = (first ? 0 : b_prev) + a
//   c[n]   = softmax_n(b_new)
// ---------------------------------------------------------------------------
__global__ __launch_bounds__(256) void routing_logits_kernel(
    const float* __restrict__ u,       // [B, J, 160]
    const float* __restrict__ vprev,   // [B, 160]
    const float* bprev,                // [B, J, 10]  (may alias bnew)
    float* bnew,                       // [B, J, 10]
    float* __restrict__ cmat,          // [B, J, 10]
    int first, int J) {
  const int idx = blockIdx.x * blockDim.x + threadIdx.x;
  const int b = idx / J;
  const int j = idx % J;
  if (b >= BATCH) return;

  const float* urow = u + ((size_t)b * J + j) * KTOT;
  const float* vp   = vprev + b * KTOT;
  const size_t bofs = ((size_t)b * J + j) * NUM_CAPS;

  float logits[NUM_CAPS];
  float mx = -3.4e38f;
#pragma unroll
  for (int n = 0; n < NUM_CAPS; ++n) {
    float dot = 0.f;
#pragma unroll
    for (int t = 0; t < CAP_LEN; ++t)
      dot = fmaf(urow[n * CAP_LEN + t], vp[n * CAP_LEN + t], dot);
    float bp = first ? 0.f : bprev[bofs + n];
    float bl = bp + dot;
    bnew[bofs + n] = bl;
    logits[n] = bl;
    mx = fmaxf(mx, bl);
  }
  float denom = 0.f;
#pragma unroll
  for (int n = 0; n < NUM_CAPS; ++n) {
    logits[n] = __expf(logits[n] - mx);
    denom += logits[n];
  }
  const float inv = 1.0f / denom;
#pragma unroll
  for (int n = 0; n < NUM_CAPS; ++n) cmat[bofs + n] = logits[n] * inv;
}

// ---------------------------------------------------------------------------
// Kernel 3: chunked deterministic reduction  s[b,k] = sum_j c[b,j,k/16]*u[b,j,k]
// cmat == nullptr encodes the first iteration's uniform c = 1/NUM_CAPS.
// blockDim = 160 (5 waves); grid = (nchunk, B).
// ---------------------------------------------------------------------------
__global__ __launch_bounds__(160) void reduce_s_kernel(
    const float* __restrict__ u,       // [B, J, 160]
    const float* __restrict__ cmat,    // [B, J, 10] or nullptr
    float* __restrict__ partials,      // [B, nchunk, 160]
    int J, int jchunk) {
  const int b  = blockIdx.y;
  const int ch = blockIdx.x;
  const int t  = threadIdx.x;          // 0..159
  const int n  = t >> 4;               // capsule of component t
  const int j0 = ch * jchunk;

  float acc = 0.f;
  for (int jj = 0; jj < jchunk; ++jj) {
    const size_t row = (size_t)b * J + (j0 + jj);
    const float  c   = cmat ? cmat[row * NUM_CAPS + n] : (1.0f / NUM_CAPS);
    acc = fmaf(c, u[row * KTOT + t], acc);
  }
  partials[((size_t)b * gridDim.x + ch) * KTOT + t] = acc;
}

__global__ void finalize_s_kernel(const float* __restrict__ partials,
                                  int nchunk, float* __restrict__ s) {
  const int i = blockIdx.x * blockDim.x + threadIdx.x;  // b*160 + t
  if (i >= BATCH * KTOT) return;
  const int b = i / KTOT, t = i % KTOT;
  float acc = 0.f;
  for (int ch = 0; ch < nchunk; ++ch)
    acc += partials[((size_t)b * nchunk + ch) * KTOT + t];
  s[i] = acc;
}

// ---------------------------------------------------------------------------
// Kernel 4: squash (elementwise -- reference's dim=1 has size 1):
//   s += bias;  n=|s|;  v = n^2/(1+n^2)/(n+eps) * s
// ---------------------------------------------------------------------------
__global__ void squash_kernel(const float* __restrict__ s_in,
                              const float* __restrict__ biases,  // [160]
                              float* __restrict__ v_out) {
  const int i = blockIdx.x * blockDim.x + threadIdx.x;
  if (i >= BATCH * KTOT) return;
  const float s  = s_in[i] + biases[i % KTOT];
  const float n  = fabsf(s);
  const float n2 = n * n;
  v_out[i] = (n2 / (1.0f + n2)) / (n + SQ_EPS) * s;
}

// ---------------------------------------------------------------------------
extern "C" void kernel_launch(void* const* d_in, const int* in_sizes, int n_in,
                              void* d_out, int out_size, void* d_ws, size_t ws_size,
                              hipStream_t stream) {
  const float* x      = (const float*)d_in[0];
  const float* weight = (const float*)d_in[1];
  const float* biases = (const float*)d_in[2];
  float*       out    = (float*)d_out;

  const int J = in_sizes[1] / (DIN * KTOT);   // 8192
  const int NCHUNK = 32;
  const int jchunk = J / NCHUNK;              // 256

  // workspace layout (floats)
  float* ws   = (float*)d_ws;
  float* u    = ws;                                   // B*J*160
  float* bmat = u + (size_t)BATCH * J * KTOT;         // B*J*10
  float* cmat = bmat + (size_t)BATCH * J * NUM_CAPS;  // B*J*10
  float* part = cmat + (size_t)BATCH * J * NUM_CAPS;  // B*NCHUNK*160
  float* sbuf = part + (size_t)BATCH * NCHUNK * KTOT; // B*160
  float* vbuf = sbuf + BATCH * KTOT;                  // B*160

  const int SM_BLK = (BATCH * KTOT + 255) / 256;      // squash/finalize grid

  // 1) pose projection (WMMA)
  pose_wmma_kernel<<<J, 320, 0, stream>>>(x, weight, u, J);

  // 2) r=0: uniform c -> s0 -> v0
  reduce_s_kernel<<<dim3(NCHUNK, BATCH), 160, 0, stream>>>(u, nullptr, part, J, jchunk);
  finalize_s_kernel<<<SM_BLK, 256, 0, stream>>>(part, NCHUNK, sbuf);
  squash_kernel<<<SM_BLK, 256, 0, stream>>>(sbuf, biases, vbuf);

  // 3) r=1: b1 = agreement(v0), c=softmax(b1) -> s1 -> v1
  routing_logits_kernel<<<(BATCH * J) / 256, 256, 0, stream>>>(
      u, vbuf, bmat, bmat, cmat, /*first=*/1, J);
  reduce_s_kernel<<<dim3(NCHUNK, BATCH), 160, 0, stream>>>(u, cmat, part, J, jchunk);
  finalize_s_kernel<<<SM_BLK, 256, 0, stream>>>(part, NCHUNK, sbuf);
  squash_kernel<<<SM_BLK, 256, 0, stream>>>(sbuf, biases, vbuf);

  // 4) r=2: b2 = b1 + agreement(v1), c=softmax(b2) -> s2 -> v2 = output
  routing_logits_kernel<<<(BATCH * J) / 256, 256, 0, stream>>>(
      u, vbuf, bmat, bmat, cmat, /*first=*/0, J);
  reduce_s_kernel<<<dim3(NCHUNK, BATCH), 160, 0, stream>>>(u, cmat, part, J, jchunk);
  finalize_s_kernel<<<SM_BLK, 256, 0, stream>>>(part, NCHUNK, sbuf);
  squash_kernel<<<SM_BLK, 256, 0, stream>>>(sbuf, biases, out);
}